// DCPGenerator_7095285973485
// MI455X (gfx1250) — compile-verified
//
#include <hip/hip_runtime.h>
#include <hip/hip_bf16.h>

// ---------------------------------------------------------------------------
// DCP dehaze (dark-channel prior + guided filter) for [16,3,512,512] f32.
//
// Box filters (r=40, zero-pad) are banded-ones matmuls on the CDNA5 matrix
// pipe via v_wmma_f32_16x16x4_f32 (f32 WMMA -> exact 6561-term sums).
//   Horizontal: OUT = IN x Band   (A=data from LDS strip, B=band)
//   Vertical:   OUT = Band x IN   (A=band, B=data, coalesced global rows)
// Horizontal passes stage a 16x224 strip per map into LDS cooperatively
// (global_load_async_to_lds_b128 when available), zero-padding the borders so
// the 7-ktile band loop fully unrolls with constant ds offsets and no range
// checks. Padded LDS row stride (228 floats) keeps the 16-lane A-fragment
// reads bank-conflict-free.
//
// Memory-bound pipeline (~hundreds of MB at 23.3 TB/s; intermediates fit L2):
//   prep -> min-pool(15,pad=1) -> dark -> hist/threshold -> A (deterministic
//   tree reductions) -> dark(img/A) -> p -> WMMA box passes of {I,p,Ip,II}
//   with fused a,b -> WMMA box passes of {a,b} with fused T and recombine.
// Workspace: 8 maps * 16 MiB = 128 MiB + ~0.4 MiB control area.
// ---------------------------------------------------------------------------

#define NIMG 16
#define HH   512
#define WW   512
#define HW   (HH * WW)          // 262144
#define NHW  (NIMG * HW)        // 4194304
#define RBOX 40
#define RMIN 7
#define TOPN 2621               // int(0.01*512*512)
#define OMEGA 0.95f
#define EPSGF 1.0e-3f

#define STRIPF 224              // 8 tiles * 16 + 2*48 band apron
#define STRIP4 56               // STRIPF / 4
#define LDSW   228              // padded LDS row stride (floats)

typedef __attribute__((ext_vector_type(2))) float v2f;
typedef __attribute__((ext_vector_type(8))) float v8f;
typedef __attribute__((ext_vector_type(4))) int   v4i;

__device__ __forceinline__ v8f wmma4(v2f a, v2f b, v8f c) {
  return __builtin_amdgcn_wmma_f32_16x16x4_f32(false, a, false, b, (short)0, c,
                                               false, false);
}

__device__ __forceinline__ float band40(int d) {
  unsigned u = (unsigned)(d + RBOX);
  return (u <= (unsigned)(2 * RBOX)) ? 1.0f : 0.0f;
}

__device__ __forceinline__ float wincnt(int i) {
  int lo = i - RBOX; if (lo < 0) lo = 0;
  int hi = i + RBOX; if (hi > 511) hi = 511;
  return (float)(hi - lo + 1);
}

// ---- async global->LDS staging (CDNA5 async path), with fallback ----
#if defined(__gfx1250__) && __has_builtin(__builtin_amdgcn_global_load_async_to_lds_b128)
#define HAS_ASYNC_LDS 1
__device__ __forceinline__ void async_ld16(const void* g, void* l) {
  __builtin_amdgcn_global_load_async_to_lds_b128(
      (__attribute__((address_space(1))) v4i*)g,
      (__attribute__((address_space(3))) v4i*)l, 0, 0);
}
__device__ __forceinline__ void async_wait0() {
#if __has_builtin(__builtin_amdgcn_s_wait_asynccnt)
  __builtin_amdgcn_s_wait_asynccnt(0);
#else
  asm volatile("s_wait_asynccnt 0" ::: "memory");
#endif
}
#endif

// Stage 16 rows x STRIPF floats of two maps into LDS, zero-padding OOB cols.
__device__ __forceinline__ void stage2(const float* __restrict__ g0,
                                       const float* __restrict__ g1,
                                       float* s0, float* s1,
                                       int base, int y0, size_t ib) {
  int tid = threadIdx.y * 32 + threadIdx.x;
  for (int i = tid; i < 16 * STRIP4; i += 256) {
    int r = i / STRIP4, c4 = i - r * STRIP4;
    int u = base + (c4 << 2);
    int lo = r * LDSW + (c4 << 2);
    const float* gr0 = g0 + ib + ((size_t)(y0 + r) << 9);
    const float* gr1 = g1 + ib + ((size_t)(y0 + r) << 9);
    if (u >= 0 && u <= WW - 4) {
#ifdef HAS_ASYNC_LDS
      async_ld16(gr0 + u, &s0[lo]);
      async_ld16(gr1 + u, &s1[lo]);
#else
      *(float4*)&s0[lo] = *(const float4*)(gr0 + u);
      *(float4*)&s1[lo] = *(const float4*)(gr1 + u);
#endif
    } else {
      #pragma unroll
      for (int k = 0; k < 4; ++k) {
        int uu = u + k;
        bool inr = (uu >= 0) && (uu < WW);
        s0[lo + k] = inr ? gr0[uu] : 0.0f;
        s1[lo + k] = inr ? gr1[uu] : 0.0f;
      }
    }
  }
#ifdef HAS_ASYNC_LDS
  async_wait0();
#endif
  __syncthreads();
}

// ---------------- elementwise prep ----------------
__global__ void prep_kernel(const float* __restrict__ x, float* __restrict__ I,
                            float* __restrict__ m) {
  int idx = blockIdx.x * 256 + threadIdx.x;
  if (idx >= NHW) return;
  int n = idx >> 18, p = idx & (HW - 1);
  const float* bx = x + (size_t)n * 3 * HW + p;
  float r = bx[0], g = bx[HW], b = bx[2 * HW];
  I[idx] = (0.2989f * r + 0.587f * g + 0.114f * b + 1.0f) * 0.5f;
  m[idx] = (fminf(r, fminf(g, b)) + 1.0f) * 0.5f;
}

// ---------------- separable 15-min with pad = 1.0 ----------------
__global__ void hmin_kernel(const float* __restrict__ in, float* __restrict__ out) {
  int idx = blockIdx.x * 256 + threadIdx.x;
  if (idx >= NHW) return;
  int col = idx & (WW - 1), rowbase = idx - col;
  float acc = 3.0e38f;
  #pragma unroll
  for (int d = -RMIN; d <= RMIN; ++d) {
    int u = col + d;
    float v = (u >= 0 && u < WW) ? in[rowbase + u] : 1.0f;
    acc = fminf(acc, v);
  }
  out[idx] = acc;
}

__global__ void vmin_kernel(const float* __restrict__ in, float* __restrict__ out,
                            int to_trans) {
  int idx = blockIdx.x * 256 + threadIdx.x;
  if (idx >= NHW) return;
  int col = idx & (WW - 1), row = (idx >> 9) & (HH - 1), base = (idx >> 18) << 18;
  float acc = 3.0e38f;
  #pragma unroll
  for (int d = -RMIN; d <= RMIN; ++d) {
    int r = row + d;
    float v = (r >= 0 && r < HH) ? in[base + (r << 9) + col] : 1.0f;
    acc = fminf(acc, v);
  }
  out[idx] = to_trans ? (1.0f - OMEGA * acc) : acc;
}

// ---------------- atmospheric light (hist -> threshold -> mean) ----------------
__global__ void zero_small_kernel(int* hist, float* part) {
  int i = blockIdx.x * 256 + threadIdx.x;
  if (i < NIMG * 1024) hist[i] = 0;
  if (i < NIMG * 1024 * 4) part[i] = 0.0f;
}

__global__ void hist_kernel(const float* __restrict__ dark, int* __restrict__ hist) {
  __shared__ int hs[1024];
  for (int i = threadIdx.x; i < 1024; i += 256) hs[i] = 0;
  __syncthreads();
  int n = blockIdx.x >> 10;
  int p = ((blockIdx.x & 1023) << 8) + threadIdx.x;
  float d = dark[(n << 18) + p];
  int b = (int)(d * 1024.0f);
  b = b < 0 ? 0 : (b > 1023 ? 1023 : b);
  atomicAdd(&hs[b], 1);                       // integer atomics: deterministic
  __syncthreads();
  for (int i = threadIdx.x; i < 1024; i += 256)
    if (hs[i]) atomicAdd(&hist[(n << 10) + i], hs[i]);
}

__global__ void thresh_kernel(const int* __restrict__ hist, float* __restrict__ thr) {
  int n = threadIdx.x;
  if (n >= NIMG) return;
  int acc = 0, b = 1023;
  for (; b > 0; --b) {
    acc += hist[(n << 10) + b];
    if (acc >= TOPN) break;
  }
  thr[n] = (float)b * (1.0f / 1024.0f);
}

__global__ void accA1_kernel(const float* __restrict__ x, const float* __restrict__ dark,
                             const float* __restrict__ thr, float4* __restrict__ part) {
  __shared__ float4 sh[256];
  int n = blockIdx.x >> 10;
  int p = ((blockIdx.x & 1023) << 8) + threadIdx.x;
  float d = dark[(n << 18) + p];
  float4 v = make_float4(0.f, 0.f, 0.f, 0.f);
  if (d >= thr[n]) {
    const float* bx = x + (size_t)n * 3 * HW + p;
    v.x = (bx[0] + 1.f) * 0.5f;
    v.y = (bx[HW] + 1.f) * 0.5f;
    v.z = (bx[2 * HW] + 1.f) * 0.5f;
    v.w = 1.0f;
  }
  sh[threadIdx.x] = v;
  __syncthreads();
  for (int s = 128; s > 0; s >>= 1) {        // fixed-order tree: deterministic
    if (threadIdx.x < s) {
      float4 o = sh[threadIdx.x + s];
      sh[threadIdx.x].x += o.x; sh[threadIdx.x].y += o.y;
      sh[threadIdx.x].z += o.z; sh[threadIdx.x].w += o.w;
    }
    __syncthreads();
  }
  if (threadIdx.x == 0) part[blockIdx.x] = sh[0];
}

__global__ void accA2_kernel(const float4* __restrict__ part, float* __restrict__ A) {
  __shared__ float4 sh[256];
  int n = blockIdx.x;
  float4 acc = make_float4(0.f, 0.f, 0.f, 0.f);
  for (int i = threadIdx.x; i < 1024; i += 256) {
    float4 v = part[(n << 10) + i];
    acc.x += v.x; acc.y += v.y; acc.z += v.z; acc.w += v.w;
  }
  sh[threadIdx.x] = acc;
  __syncthreads();
  for (int s = 128; s > 0; s >>= 1) {
    if (threadIdx.x < s) {
      float4 o = sh[threadIdx.x + s];
      sh[threadIdx.x].x += o.x; sh[threadIdx.x].y += o.y;
      sh[threadIdx.x].z += o.z; sh[threadIdx.x].w += o.w;
    }
    __syncthreads();
  }
  if (threadIdx.x == 0) {
    float4 t = sh[0];
    float inv = (t.w > 0.f) ? 1.0f / t.w : 0.0f;
    A[n * 3 + 0] = t.x * inv; A[n * 3 + 1] = t.y * inv; A[n * 3 + 2] = t.z * inv;
  }
}

__global__ void m2_kernel(const float* __restrict__ x, const float* __restrict__ A,
                          float* __restrict__ m2) {
  int idx = blockIdx.x * 256 + threadIdx.x;
  if (idx >= NHW) return;
  int n = idx >> 18, p = idx & (HW - 1);
  const float* bx = x + (size_t)n * 3 * HW + p;
  float v0 = (bx[0] + 1.f) * 0.5f / A[n * 3 + 0];
  float v1 = (bx[HW] + 1.f) * 0.5f / A[n * 3 + 1];
  float v2 = (bx[2 * HW] + 1.f) * 0.5f / A[n * 3 + 2];
  m2[idx] = fminf(v0, fminf(v1, v2));
}

// ---------------- WMMA horizontal box-sum (LDS-staged) ----------------
// grid (4, NIMG*32), block (32,8): block = one y-tile row, 8 x-tiles.
// Wave w owns x-tile tx = bx*8+w; its band inputs live at strip-local
// offsets [w*16, w*16+111], all inside the zero-padded LDS strip.

__global__ __launch_bounds__(256)
void hbox4_kernel(const float* __restrict__ I, const float* __restrict__ p,
                  float* __restrict__ tI, float* __restrict__ tp,
                  float* __restrict__ tIp, float* __restrict__ tII) {
  __shared__ float sI[16 * LDSW];
  __shared__ float sP[16 * LDSW];
  int lane = threadIdx.x, wid = threadIdx.y;
  int Mi = lane & 15, half = lane >> 4;
  int bx = blockIdx.x;
  int tyAll = blockIdx.y;
  int n = tyAll >> 5, ty = tyAll & 31;
  int y0 = ty << 4;
  size_t ib = (size_t)n << 18;
  int base = (bx << 7) - 48;

  stage2(I, p, sI, sP, base, y0, ib);

  const float* rI = &sI[Mi * LDSW + (wid << 4)];
  const float* rp = &sP[Mi * LDSW + (wid << 4)];
  v8f cI = {}, cp = {}, cIp = {}, cII = {};
  #pragma unroll
  for (int kt = 0; kt < 7; ++kt) {
    #pragma unroll
    for (int kc = 0; kc < 4; ++kc) {
      int uo = (kt << 4) + (kc << 2) + (half << 1);
      float i0 = rI[uo], i1 = rI[uo + 1];
      float p0 = rp[uo], p1 = rp[uo + 1];
      int d0 = 48 + Mi - uo;                // col - u, block-independent
      v2f b; b.x = band40(d0); b.y = band40(d0 - 1);
      v2f a;
      a.x = i0;      a.y = i1;      cI  = wmma4(a, b, cI);
      a.x = p0;      a.y = p1;      cp  = wmma4(a, b, cp);
      a.x = i0 * p0; a.y = i1 * p1; cIp = wmma4(a, b, cIp);
      a.x = i0 * i0; a.y = i1 * i1; cII = wmma4(a, b, cII);
    }
  }
  int col = (bx << 7) + (wid << 4) + Mi;
  #pragma unroll
  for (int v = 0; v < 8; ++v) {
    size_t o = ib + ((size_t)(y0 + v + (half << 3)) << 9) + col;
    tI[o] = cI[v]; tp[o] = cp[v]; tIp[o] = cIp[v]; tII[o] = cII[v];
  }
}

__global__ __launch_bounds__(256)
void hbox2_kernel(const float* __restrict__ amap, const float* __restrict__ bmap,
                  float* __restrict__ ta, float* __restrict__ tb) {
  __shared__ float sA[16 * LDSW];
  __shared__ float sB[16 * LDSW];
  int lane = threadIdx.x, wid = threadIdx.y;
  int Mi = lane & 15, half = lane >> 4;
  int bx = blockIdx.x;
  int tyAll = blockIdx.y;
  int n = tyAll >> 5, ty = tyAll & 31;
  int y0 = ty << 4;
  size_t ib = (size_t)n << 18;
  int base = (bx << 7) - 48;

  stage2(amap, bmap, sA, sB, base, y0, ib);

  const float* ra = &sA[Mi * LDSW + (wid << 4)];
  const float* rb = &sB[Mi * LDSW + (wid << 4)];
  v8f ca = {}, cb = {};
  #pragma unroll
  for (int kt = 0; kt < 7; ++kt) {
    #pragma unroll
    for (int kc = 0; kc < 4; ++kc) {
      int uo = (kt << 4) + (kc << 2) + (half << 1);
      int d0 = 48 + Mi - uo;
      v2f b; b.x = band40(d0); b.y = band40(d0 - 1);
      v2f a;
      a.x = ra[uo]; a.y = ra[uo + 1]; ca = wmma4(a, b, ca);
      a.x = rb[uo]; a.y = rb[uo + 1]; cb = wmma4(a, b, cb);
    }
  }
  int col = (bx << 7) + (wid << 4) + Mi;
  #pragma unroll
  for (int v = 0; v < 8; ++v) {
    size_t o = ib + ((size_t)(y0 + v + (half << 3)) << 9) + col;
    ta[o] = ca[v]; tb[o] = cb[v];
  }
}

// ---------------- WMMA vertical box-sum passes ----------------
// block (32,8), grid (32,64): wave = one 16x16 tile; B loads are coalesced
// 64B row segments (L2-resident intermediates).
__device__ __forceinline__ void tile_ids(int& n, int& ty, int& tx, int& Mi, int& half) {
  int lane = threadIdx.x;
  Mi = lane & 15; half = lane >> 4;
  tx = blockIdx.x;
  int tyAll = blockIdx.y * blockDim.y + threadIdx.y;
  n = tyAll >> 5; ty = tyAll & 31;
}

__global__ __launch_bounds__(256)
void vbox4ab_kernel(const float* __restrict__ tI, const float* __restrict__ tp,
                    const float* __restrict__ tIp, const float* __restrict__ tII,
                    float* __restrict__ amap, float* __restrict__ bmap) {
  int n, ty, tx, Mi, half; tile_ids(n, ty, tx, Mi, half);
  size_t ib = (size_t)n << 18;
  int x0 = tx << 4, y0 = ty << 4;
  int col = x0 + Mi, orowA = y0 + Mi;
  v8f cI = {}, cp = {}, cIp = {}, cII = {};
  int klo = ty - 3; if (klo < 0) klo = 0;
  int khi = ty + 3; if (khi > 31) khi = 31;
  for (int kt = klo; kt <= khi; ++kt) {
    int vb = kt << 4;
    #pragma unroll
    for (int kc = 0; kc < 4; ++kc) {
      int v0 = vb + (kc << 2) + (half << 1);
      v2f a; a.x = band40(orowA - v0); a.y = band40(orowA - (v0 + 1));
      size_t r0 = ib + ((size_t)v0 << 9) + col, r1 = r0 + WW;
      v2f b;
      b.x = tI[r0];  b.y = tI[r1];  cI  = wmma4(a, b, cI);
      b.x = tp[r0];  b.y = tp[r1];  cp  = wmma4(a, b, cp);
      b.x = tIp[r0]; b.y = tIp[r1]; cIp = wmma4(a, b, cIp);
      b.x = tII[r0]; b.y = tII[r1]; cII = wmma4(a, b, cII);
    }
  }
  #pragma unroll
  for (int v = 0; v < 8; ++v) {
    int orow = y0 + v + (half << 3);
    float invc = 1.0f / (wincnt(orow) * wincnt(col));
    float mI = cI[v] * invc, mp = cp[v] * invc;
    float cov = cIp[v] * invc - mI * mp;
    float var = cII[v] * invc - mI * mI;
    float aa = cov / (var + EPSGF);
    float bb = mp - aa * mI;
    size_t o = ib + ((size_t)orow << 9) + col;
    amap[o] = aa; bmap[o] = bb;
  }
}

__global__ __launch_bounds__(256)
void vbox2fin_kernel(const float* __restrict__ ta, const float* __restrict__ tb,
                     const float* __restrict__ I, const float* __restrict__ x,
                     const float* __restrict__ A, float* __restrict__ out) {
  int n, ty, tx, Mi, half; tile_ids(n, ty, tx, Mi, half);
  size_t ib = (size_t)n << 18;
  int x0 = tx << 4, y0 = ty << 4;
  int col = x0 + Mi, orowA = y0 + Mi;
  v8f ca = {}, cb = {};
  int klo = ty - 3; if (klo < 0) klo = 0;
  int khi = ty + 3; if (khi > 31) khi = 31;
  for (int kt = klo; kt <= khi; ++kt) {
    int vb = kt << 4;
    #pragma unroll
    for (int kc = 0; kc < 4; ++kc) {
      int v0 = vb + (kc << 2) + (half << 1);
      v2f a; a.x = band40(orowA - v0); a.y = band40(orowA - (v0 + 1));
      size_t r0 = ib + ((size_t)v0 << 9) + col, r1 = r0 + WW;
      v2f b;
      b.x = ta[r0]; b.y = ta[r1]; ca = wmma4(a, b, ca);
      b.x = tb[r0]; b.y = tb[r1]; cb = wmma4(a, b, cb);
    }
  }
  float A0 = A[n * 3 + 0], A1 = A[n * 3 + 1], A2 = A[n * 3 + 2];
  #pragma unroll
  for (int v = 0; v < 8; ++v) {
    int orow = y0 + v + (half << 3);
    float invc = 1.0f / (wincnt(orow) * wincnt(col));
    size_t o = ib + ((size_t)orow << 9) + col;
    float T = ca[v] * invc * I[o] + cb[v] * invc;
    float invT = 1.0f / T;
    size_t xo = (size_t)n * 3 * HW + ((size_t)orow << 9) + col;
    float i0 = (x[xo] + 1.f) * 0.5f;
    float i1 = (x[xo + HW] + 1.f) * 0.5f;
    float i2 = (x[xo + 2 * HW] + 1.f) * 0.5f;
    out[xo]          = (i0 - A0) * invT + A0;
    out[xo + HW]     = (i1 - A1) * invT + A1;
    out[xo + 2 * HW] = (i2 - A2) * invT + A2;
  }
}

// ---------------------------------------------------------------------------
extern "C" void kernel_launch(void* const* d_in, const int* in_sizes, int n_in,
                              void* d_out, int out_size, void* d_ws, size_t ws_size,
                              hipStream_t stream) {
  (void)in_sizes; (void)n_in; (void)out_size; (void)ws_size;
  const float* x = (const float*)d_in[0];
  float* out = (float*)d_out;

  const size_t MAPB = (size_t)NHW * sizeof(float);  // 16 MiB
  float* s0 = (float*)d_ws;                         // I (guidance)
  float* s1 = s0 + (size_t)NHW * 1;
  float* s2 = s0 + (size_t)NHW * 2;
  float* s3 = s0 + (size_t)NHW * 3;
  float* s4 = s0 + (size_t)NHW * 4;
  float* s5 = s0 + (size_t)NHW * 5;
  float* s6 = s0 + (size_t)NHW * 6;
  float* s7 = s0 + (size_t)NHW * 7;
  char* small = (char*)d_ws + 8 * MAPB;
  int*    hist = (int*)small;                                  // 64 KiB
  float*  part = (float*)(small + 65536);                      // 256 KiB
  float*  thr  = (float*)(small + 65536 + 262144);
  float*  Aatm = (float*)(small + 65536 + 262144 + 256);

  const int EB = 256, EG = NHW / 256;
  dim3 bW(32, 8);
  dim3 gH(WW / 128, NIMG * HH / 16);                // hbox: (4, 512)
  dim3 gV(WW / 16, (NIMG * HH / 16) / 8);           // vbox: (32, 64)

  zero_small_kernel<<<NIMG * 1024 * 4 / 256, 256, 0, stream>>>(hist, part);
  prep_kernel<<<EG, EB, 0, stream>>>(x, s0, s1);               // I, m
  hmin_kernel<<<EG, EB, 0, stream>>>(s1, s2);
  vmin_kernel<<<EG, EB, 0, stream>>>(s2, s3, 0);               // dark
  hist_kernel<<<NIMG * 1024, 256, 0, stream>>>(s3, hist);
  thresh_kernel<<<1, 32, 0, stream>>>(hist, thr);
  accA1_kernel<<<NIMG * 1024, 256, 0, stream>>>(x, s3, thr, (float4*)part);
  accA2_kernel<<<NIMG, 256, 0, stream>>>((const float4*)part, Aatm);
  m2_kernel<<<EG, EB, 0, stream>>>(x, Aatm, s3);               // min_c img/A
  hmin_kernel<<<EG, EB, 0, stream>>>(s3, s2);
  vmin_kernel<<<EG, EB, 0, stream>>>(s2, s4, 1);               // p
  hbox4_kernel<<<gH, bW, 0, stream>>>(s0, s4, s1, s2, s3, s5);
  vbox4ab_kernel<<<gV, bW, 0, stream>>>(s1, s2, s3, s5, s6, s7);
  hbox2_kernel<<<gH, bW, 0, stream>>>(s6, s7, s1, s2);
  vbox2fin_kernel<<<gV, bW, 0, stream>>>(s1, s2, s0, x, Aatm, out);
}